// SpatialAttention_24610162606461
// MI455X (gfx1250) — compile-verified
//
#include <hip/hip_runtime.h>
#include <hip/hip_bf16.h>
#include <math.h>

// ---------------------------------------------------------------------------
// SpatialAttention: pooled = mean_c(x) + max_c(x); attn = sigmoid(conv7x7(pooled)+b)
// out = x * attn  (broadcast over channels)
// x: [32, 256, 64, 64] fp32.  Bandwidth-bound: ~384MB min HBM traffic @23.3TB/s.
// CDNA5 paths: TDM (tensor_load_to_lds) stages attn tiles into LDS;
//              conv7x7 runs on the matrix pipe via V_WMMA_F32_16X16X4_F32.
// ---------------------------------------------------------------------------

#define B_   32
#define C_   256
#define HW_  4096          // 64*64
#define H_   64
#define W_   64

typedef unsigned int v4u __attribute__((ext_vector_type(4)));
typedef int          v8i __attribute__((ext_vector_type(8)));
typedef int          v4i __attribute__((ext_vector_type(4)));
typedef float        v2f __attribute__((ext_vector_type(2)));
typedef float        v8f __attribute__((ext_vector_type(8)));

#if defined(__has_builtin)
# if __has_builtin(__builtin_amdgcn_tensor_load_to_lds) && __has_builtin(__builtin_amdgcn_s_wait_tensorcnt)
#  define HAS_TDM 1
# endif
# if __has_builtin(__builtin_amdgcn_wmma_f32_16x16x4_f32)
#  define HAS_WMMA 1
# endif
#endif
#ifndef HAS_TDM
# define HAS_TDM 0
#endif
#ifndef HAS_WMMA
# define HAS_WMMA 0
#endif

#if HAS_TDM
// Issue one 1-D TDM transfer: n_elems fp32 from gsrc -> LDS byte offset lds_off.
// D# layout per CDNA5 ISA ch.8 (group0: count/lds_addr/global_addr/type,
// group1: data_size, tensor_dim0/1, tile_dim0/1, stride0). 6-arg builtin form.
__device__ __forceinline__ void tdm_load_1d_f32(const float* gsrc,
                                                unsigned lds_off,
                                                unsigned n_elems) {
  unsigned long long ga = (unsigned long long)(size_t)gsrc;
  v4u g0;
  g0[0] = 1u;                                         // count=1, user mode
  g0[1] = lds_off;                                    // lds_addr [63:32]
  g0[2] = (unsigned)(ga & 0xFFFFFFFFu);               // global_addr[31:0]
  g0[3] = (unsigned)((ga >> 32) & 0x01FFFFFFu)        // global_addr[56:32]
        | (2u << 30);                                 // type=2 ("image")
  v8i g1;
  g1[0] = (int)(2u << 16);                            // data_size=4B, mask=0
  g1[1] = (int)((n_elems & 0xFFFFu) << 16);           // tensor_dim0[15:0] @ [63:48]
  g1[2] = (int)((n_elems >> 16) | (1u << 16));        // dim0[31:16] | tensor_dim1=1
  g1[3] = (int)((n_elems & 0xFFFFu) << 16);           // tile_dim0 @ [127:112]
  g1[4] = 1;                                          // tile_dim1=1, tile_dim2=0
  g1[5] = (int)n_elems;                               // tensor_dim0_stride[31:0]
  g1[6] = 0;
  g1[7] = 0;
  v4i z4 = {0, 0, 0, 0};
  v8i z8 = {0, 0, 0, 0, 0, 0, 0, 0};
  __builtin_amdgcn_tensor_load_to_lds(g0, g1, z4, z4, z8, 0);
}
#endif

// ---------------------------------------------------------------------------
// Kernel 1: channel pooling. One thread per (b, spatial) pixel.
// Adjacent threads -> adjacent spatial index: each channel-loop iteration is a
// fully coalesced 128B line per wave; unroll 16 -> deep s_clause load groups
// (~8MB in flight across the grid, enough to cover HBM latency at 23.3TB/s).
// ---------------------------------------------------------------------------
__global__ void sa_pool_kernel(const float* __restrict__ x,
                               float* __restrict__ pooled) {
  int p = blockIdx.x * blockDim.x + threadIdx.x;      // [0, 32*4096)
  int b = p >> 12;
  int s = p & (HW_ - 1);
  const float* xp = x + (size_t)b * C_ * HW_ + s;
  float sum = 0.0f;
  float mx  = -3.402823466e+38f;
  #pragma unroll 16
  for (int c = 0; c < C_; ++c) {
    float v = xp[(size_t)c * HW_];
    sum += v;
    mx = fmaxf(mx, v);
  }
  pooled[p] = sum * (1.0f / (float)C_) + mx;
}

// ---------------------------------------------------------------------------
// Kernel 2 (WMMA): 7x7 conv + bias + sigmoid on the matrix pipe.
// Each wave computes 16 consecutive pixels (same row): D[16x16] accumulated by
// 13 x V_WMMA_F32_16X16X4_F32 over K=52 taps (49 real + 3 zero pad).
//   A (16x4 f32, 2 VGPRs): lane L holds patch(M=L&15, k=4t+2*(L>>4)+v)
//   B (4x16 f32, 2 VGPRs): row k striped across lanes; w[k] replicated over N
//   => D[m][n] = conv(m) for every n; lanes with N==0 write 8 results each.
// Loads use clamped addresses + select (no divergence -> EXEC all-1 at WMMA).
// ---------------------------------------------------------------------------
#if HAS_WMMA
__device__ __forceinline__ float sa_patch_val(const float* __restrict__ pl,
                                              int y, int xc, int k) {
  // tap k in [0,52); zero for k>=49 or out-of-image
  int ky = k / 7;
  int kx = k - ky * 7;
  int yy = y + ky - 3;
  int xx = xc + kx - 3;
  bool inb = (k < 49) & ((unsigned)yy < (unsigned)H_) & ((unsigned)xx < (unsigned)W_);
  int cy = min(max(yy, 0), H_ - 1);
  int cx = min(max(xx, 0), W_ - 1);
  float v = pl[cy * W_ + cx];                         // always-in-bounds load
  return inb ? v : 0.0f;
}

__global__ void sa_conv_wmma_kernel(const float* __restrict__ pooled,
                                    const float* __restrict__ w,
                                    const float* __restrict__ bias,
                                    float* __restrict__ attn) {
  int tid    = threadIdx.x;
  int lane   = tid & 31;
  int waveid = tid >> 5;
  int pixbase = (blockIdx.x * 8 + waveid) << 4;       // 16 pixels per wave
  int p  = pixbase + (lane & 15);                     // this lane's M pixel
  int b  = p >> 12;
  int y  = (p >> 6) & (H_ - 1);
  int xc = p & (W_ - 1);
  const float* pl = pooled + ((size_t)b << 12);

  int khalf = (lane >> 4) << 1;                       // K offset for this half-wave
  v8f acc = {0.f, 0.f, 0.f, 0.f, 0.f, 0.f, 0.f, 0.f};

  #pragma unroll
  for (int t = 0; t < 13; ++t) {
    int k0 = 4 * t + khalf;
    v2f A, Bm;
    A[0]  = sa_patch_val(pl, y, xc, k0);
    A[1]  = sa_patch_val(pl, y, xc, k0 + 1);
    Bm[0] = (k0     < 49) ? w[k0]     : 0.0f;
    Bm[1] = (k0 + 1 < 49) ? w[k0 + 1] : 0.0f;
    acc = __builtin_amdgcn_wmma_f32_16x16x4_f32(
        /*neg_a=*/false, A, /*neg_b=*/false, Bm,
        /*c_mod=*/(short)0, acc, /*reuse_a=*/false, /*reuse_b=*/false);
  }

  // D layout: VGPR r, lane L -> M = r + 8*(L>>4), N = L&15. Take N==0 lanes.
  float bv = bias[0];
  if ((lane & 15) == 0) {
    int mbase = pixbase + ((lane >> 4) << 3);
    #pragma unroll
    for (int r = 0; r < 8; ++r)
      attn[mbase + r] = 1.0f / (1.0f + __expf(-(acc[r] + bv)));
  }
}
#endif

// ---------------------------------------------------------------------------
// Kernel 2 (fallback, scalar VALU conv) — used only if WMMA builtin absent.
// ---------------------------------------------------------------------------
__global__ void sa_conv_kernel(const float* __restrict__ pooled,
                               const float* __restrict__ w,
                               const float* __restrict__ bias,
                               float* __restrict__ attn) {
  int p  = blockIdx.x * blockDim.x + threadIdx.x;     // [0, 32*4096)
  int b  = p >> 12;
  int y  = (p >> 6) & (H_ - 1);
  int xc = p & (W_ - 1);
  const float* pl = pooled + ((size_t)b << 12);
  float acc = bias[0];
  #pragma unroll
  for (int ky = 0; ky < 7; ++ky) {
    int yy = y + ky - 3;
    if ((unsigned)yy < (unsigned)H_) {
      #pragma unroll
      for (int kx = 0; kx < 7; ++kx) {
        int xx = xc + kx - 3;
        if ((unsigned)xx < (unsigned)W_)
          acc = fmaf(pl[yy * W_ + xx], w[ky * 7 + kx], acc);
      }
    }
  }
  attn[p] = 1.0f / (1.0f + __expf(-acc));
}

// ---------------------------------------------------------------------------
// Kernel 3: out = x * attn (broadcast over channels).
// Block = (batch b, 1024-pixel span j, 16-channel group g). The 4KB attn span
// is DMA'd into LDS once via the Tensor Data Mover, then reused for 16
// channels of float4 streaming multiplies (B128 loads/stores, fully coalesced).
// ---------------------------------------------------------------------------
__global__ void sa_scale_kernel(const float* __restrict__ x,
                                const float* __restrict__ attn,
                                float* __restrict__ out) {
  __shared__ float sm[1024];

  int bid = blockIdx.x;            // [0, 32*4*16)
  int b   = bid >> 6;
  int rem = bid & 63;
  int j   = rem >> 4;              // span [0,4): 1024 pixels each
  int g   = rem & 15;              // channel group [0,16): 16 channels each
  int tid = threadIdx.x;           // [0,256)

  int span_off = j << 10;          // pixel offset within plane
  const float* asrc = attn + ((size_t)b << 12) + span_off;

#if HAS_TDM
  if (tid == 0) {
    unsigned lds_off = (unsigned)(size_t)(&sm[0]);    // low 32b of flat = LDS offset
    tdm_load_1d_f32(asrc, lds_off, 1024u);
    __builtin_amdgcn_s_wait_tensorcnt(0);
  }
  __syncthreads();
#else
  reinterpret_cast<float4*>(sm)[tid] =
      reinterpret_cast<const float4*>(asrc)[tid];
  __syncthreads();
#endif

  const float4 av = *reinterpret_cast<const float4*>(&sm[tid << 2]);

  int c0 = g << 4;
  size_t base = ((size_t)b * C_ + c0) * HW_ + span_off + (tid << 2);
  #pragma unroll 4
  for (int c = 0; c < 16; ++c) {
    size_t idx = base + (size_t)c * HW_;
    float4 xv = *reinterpret_cast<const float4*>(x + idx);
    float4 ov;
    ov.x = xv.x * av.x;
    ov.y = xv.y * av.y;
    ov.z = xv.z * av.z;
    ov.w = xv.w * av.w;
    *reinterpret_cast<float4*>(out + idx) = ov;
  }
}

// ---------------------------------------------------------------------------
extern "C" void kernel_launch(void* const* d_in, const int* in_sizes, int n_in,
                              void* d_out, int out_size, void* d_ws, size_t ws_size,
                              hipStream_t stream) {
  const float* x    = (const float*)d_in[0];   // 32*256*64*64
  const float* w    = (const float*)d_in[1];   // 49
  const float* bias = (const float*)d_in[2];   // 1
  float* out = (float*)d_out;

  float* pooled = (float*)d_ws;                // 131072 floats (512KB)
  float* attn   = pooled + (B_ * HW_);         // 131072 floats (512KB)

  // Kernel 1: pooling. 32*4096 = 131072 threads.
  sa_pool_kernel<<<(B_ * HW_) / 256, 256, 0, stream>>>(x, pooled);

  // Kernel 2: conv + sigmoid on the matrix pipe (16 px/wave, 8 waves/block).
#if HAS_WMMA
  sa_conv_wmma_kernel<<<(B_ * HW_) / 128, 256, 0, stream>>>(pooled, w, bias, attn);
#else
  sa_conv_kernel<<<(B_ * HW_) / 256, 256, 0, stream>>>(pooled, w, bias, attn);
#endif

  // Kernel 3: broadcast multiply. 32 batches * 4 spans * 16 channel-groups.
  sa_scale_kernel<<<B_ * 4 * 16, 256, 0, stream>>>(x, attn, out);
}